// WeightedTensorProduct_1030792151279
// MI455X (gfx1250) — compile-verified
//
#include <hip/hip_runtime.h>

// CDNA5 / gfx1250, wave32. One wave per batch element.
// out[z, wo*(2lo+1)+k] = inv[lo] * sum_paths W_path[z,wo,u] * t_path[z,u,k]
// t computed per-lane in B-fragment layout; W loaded per-lane in A-fragment
// layout; contraction via V_WMMA_F32_16X16X4_F32 (4 chained K=4 steps).

typedef __attribute__((ext_vector_type(2))) float v2f;
typedef __attribute__((ext_vector_type(8))) float v8f;

#define DEV __device__ __forceinline__

struct Coef { int k, i, j; float v; };   // out-index, x1-index (in l1 row), absolute x2-index, value

// ---- Wigner 3j sparse tables (real SH basis, e3nn 'component' norm = unit-Frobenius * sqrt(2lo+1)) ----
#define S2   0.70710678f   // 1/sqrt(2)
#define Q6   0.40824829f   // 1/sqrt(6)
#define Q6B  0.81649658f   // 2/sqrt(6)
#define C3   0.57735027f   // 1/sqrt(3)
#define C5   0.44721360f   // 1/sqrt(5)
#define T1   0.54772256f   // sqrt(3/10)
#define T2   0.31622777f   // 1/sqrt(10)
#define T3   0.63245553f   // 2/sqrt(10)
#define UA   0.53452248f   // 2/sqrt(14)
#define UB   0.26726124f   // 1/sqrt(14)
#define UC   0.46291005f   // sqrt(3/14)

constexpr Coef P0[]  = {{0,0,0,1.0f}};                                                        // (l1,l2,lo)=(0,0,0)
constexpr Coef P1[]  = {{0,0,1,1.0f},{1,0,2,1.0f},{2,0,3,1.0f}};                              // (0,1,1)
constexpr Coef P2[]  = {{0,0,4,1.0f},{1,0,5,1.0f},{2,0,6,1.0f},{3,0,7,1.0f},{4,0,8,1.0f}};    // (0,2,2)
constexpr Coef P3[]  = {{0,0,0,1.0f},{1,1,0,1.0f},{2,2,0,1.0f}};                              // (1,0,1)
constexpr Coef P4[]  = {{0,0,1,C3},{0,1,2,C3},{0,2,3,C3}};                                    // (1,1,0)
constexpr Coef P5[]  = {{0,1,3,S2},{0,2,2,-S2},{1,2,1,S2},{1,0,3,-S2},{2,0,2,S2},{2,1,1,-S2}};// (1,1,1) eps/sqrt2
constexpr Coef P6[]  = {{0,0,3,S2},{0,2,1,S2},{1,0,2,S2},{1,1,1,S2},                          // (1,1,2) M-tensor
                        {2,0,1,-Q6},{2,1,2,Q6B},{2,2,3,-Q6},
                        {3,1,3,S2},{3,2,2,S2},{4,0,1,-S2},{4,2,3,S2}};
constexpr Coef P7[]  = {{0,2,4,T1},{2,0,4,T1},{0,1,5,T1},{1,0,5,T1},                          // (1,2,1)
                        {0,0,6,-T2},{1,1,6,T3},{2,2,6,-T2},
                        {1,2,7,T1},{2,1,7,T1},{0,0,8,-T1},{2,2,8,T1}};
constexpr Coef P8[]  = {{0,0,5,-Q6},{1,0,4,Q6},{0,2,7,Q6},{3,2,4,-Q6},                        // (1,2,2) T-tensor
                        {0,1,8,-Q6B},{4,1,4,Q6B},{1,2,6,S2},{2,2,5,-S2},
                        {1,1,7,-Q6},{3,1,5,Q6},{1,2,8,Q6},{4,2,5,-Q6},
                        {2,0,7,S2},{3,0,6,-S2},{3,0,8,Q6},{4,0,7,-Q6}};
constexpr Coef P9[]  = {{0,0,0,1.0f},{1,1,0,1.0f},{2,2,0,1.0f},{3,3,0,1.0f},{4,4,0,1.0f}};    // (2,0,2)
constexpr Coef P10[] = {{0,0,3,T1},{2,0,1,T1},{0,1,2,T1},{1,1,1,T1},                          // (2,1,1)
                        {0,2,1,-T2},{1,2,2,T3},{2,2,3,-T2},
                        {1,3,3,T1},{2,3,2,T1},{0,4,1,-T1},{2,4,3,T1}};
constexpr Coef P11[] = {{0,1,1,-Q6},{1,0,1,Q6},{0,3,3,Q6},{3,0,3,-Q6},                        // (2,1,2)
                        {0,4,2,-Q6B},{4,0,2,Q6B},{1,2,3,S2},{2,1,3,-S2},
                        {1,3,2,-Q6},{3,1,2,Q6},{1,4,3,Q6},{4,1,3,-Q6},
                        {2,3,1,S2},{3,2,1,-S2},{3,4,1,Q6},{4,3,1,-Q6}};
constexpr Coef P12[] = {{0,0,4,C5},{0,1,5,C5},{0,2,6,C5},{0,3,7,C5},{0,4,8,C5}};              // (2,2,0)
constexpr Coef P13[] = {{0,0,5,-T2},{0,1,4,T2},{2,0,7,T2},{2,3,4,-T2},                        // (2,2,1)
                        {1,0,8,-T3},{1,4,4,T3},{2,1,6,T1},{2,2,5,-T1},
                        {1,1,7,-T2},{1,3,5,T2},{2,1,8,T2},{2,4,5,-T2},
                        {0,2,7,T1},{0,3,6,-T1},{0,3,8,T2},{0,4,7,-T2}};
constexpr Coef P14[] = {{0,0,6,-UA},{0,2,4,-UA},{2,0,4,-UA},                                  // (2,2,2) U-tensor
                        {1,1,6,UB},{1,2,5,UB},{2,1,5,UB},
                        {3,3,6,UB},{3,2,7,UB},{2,3,7,UB},
                        {1,1,8,-UC},{1,4,5,-UC},{4,1,5,-UC},
                        {3,3,8,UC},{3,4,7,UC},{4,3,7,UC},
                        {2,2,6,UA},
                        {4,4,6,-UA},{4,2,8,-UA},{2,4,8,-UA},
                        {0,1,7,UC},{0,3,5,UC},{1,0,7,UC},{1,3,4,UC},{3,0,5,UC},{3,1,4,UC}};

// One path: build B fragments per-lane, load A (weight) fragments from global,
// accumulate 4 chained f32 16x16x4 WMMAs into acc.
template<int N, int RD>
DEV void path_wmma(const Coef (&tbl)[N], const float (&row)[8][RD],
                   const float (&x2v)[9], int n, int kb,
                   const float* __restrict__ wp, v8f& acc)
{
    // A fragments: lane holds W[m = lane&15, K = 4s+kb .. +1] -> contiguous float2
    v2f afr[4];
#pragma unroll
    for (int s = 0; s < 4; ++s)
        afr[s] = *(const v2f*)(wp + n * 16 + 4 * s + kb);

    // B fragments: lane owns column n; tac[2s+h] = t[u = 4s+kb+h, n]
    float tac[8];
#pragma unroll
    for (int m = 0; m < 8; ++m) tac[m] = 0.0f;
#pragma unroll
    for (int c = 0; c < N; ++c) {
        float cv = (n == tbl[c].k) ? (tbl[c].v * x2v[tbl[c].j]) : 0.0f;
#pragma unroll
        for (int m = 0; m < 8; ++m)
            tac[m] = fmaf(cv, row[m][tbl[c].i], tac[m]);
    }
#pragma unroll
    for (int s = 0; s < 4; ++s) {
        v2f b; b.x = tac[2 * s]; b.y = tac[2 * s + 1];
        acc = __builtin_amdgcn_wmma_f32_16x16x4_f32(
            /*neg_a=*/false, afr[s], /*neg_b=*/false, b,
            /*c_mod=*/(short)0, acc, /*reuse_a=*/false, /*reuse_b=*/false);
    }
}

__global__ __launch_bounds__(256)
void wtp_kernel(const float* __restrict__ x1, const float* __restrict__ x2,
                const float* __restrict__ w, float* __restrict__ out, int batch)
{
    __shared__ float sX1[8 * 144];
    __shared__ float sX2[8 * 9];

    const int tid  = threadIdx.x;
    const int wave = tid >> 5;
    const int lane = tid & 31;
    const int n    = lane & 15;          // output column / A row / B column
    const int kb   = (lane >> 4) << 1;   // K sub-pair: {0,1} or {2,3}

    const long long zb = (long long)blockIdx.x * 8;

    // Cooperative coalesced staging of x1/x2 for the 8 elements of this block
    {
        const int e1 = min(8 * 144, (int)((long long)batch * 144 - zb * 144));
        const float* g1 = x1 + zb * 144;
        for (int idx = tid; idx < e1; idx += 256) sX1[idx] = g1[idx];
        const int e2 = min(8 * 9, (int)((long long)batch * 9 - zb * 9));
        const float* g2 = x2 + zb * 9;
        if (tid < e2) sX2[tid] = g2[tid];
    }
    __syncthreads();

    const long long z = zb + wave;
    if (z >= batch) return;              // wave-uniform: EXEC stays all-ones for WMMA

    float x2v[9];
#pragma unroll
    for (int j = 0; j < 9; ++j) x2v[j] = sX2[wave * 9 + j];

    const float* sx = sX1 + wave * 144;
    int uu[8];
#pragma unroll
    for (int m = 0; m < 8; ++m) uu[m] = (m >> 1) * 4 + kb + (m & 1);

    v8f acc0 = {}; v8f acc1 = {}; v8f acc2 = {};
    const float* wz = w + z * 3840;

    { // l1 = 0 block (x1 offset 0)
        float r0[8][1];
#pragma unroll
        for (int m = 0; m < 8; ++m) r0[m][0] = sx[uu[m]];
        path_wmma(P0, r0, x2v, n, kb, wz + 0,    acc0);
        path_wmma(P1, r0, x2v, n, kb, wz + 256,  acc1);
        path_wmma(P2, r0, x2v, n, kb, wz + 512,  acc2);
    }
    { // l1 = 1 block (x1 offset 16)
        float r1[8][3];
#pragma unroll
        for (int m = 0; m < 8; ++m)
#pragma unroll
            for (int i = 0; i < 3; ++i) r1[m][i] = sx[16 + uu[m] * 3 + i];
        path_wmma(P3, r1, x2v, n, kb, wz + 768,  acc1);
        path_wmma(P4, r1, x2v, n, kb, wz + 1024, acc0);
        path_wmma(P5, r1, x2v, n, kb, wz + 1280, acc1);
        path_wmma(P6, r1, x2v, n, kb, wz + 1536, acc2);
        path_wmma(P7, r1, x2v, n, kb, wz + 1792, acc1);
        path_wmma(P8, r1, x2v, n, kb, wz + 2048, acc2);
    }
    { // l1 = 2 block (x1 offset 64)
        float r2[8][5];
#pragma unroll
        for (int m = 0; m < 8; ++m)
#pragma unroll
            for (int i = 0; i < 5; ++i) r2[m][i] = sx[64 + uu[m] * 5 + i];
        path_wmma(P9,  r2, x2v, n, kb, wz + 2304, acc2);
        path_wmma(P10, r2, x2v, n, kb, wz + 2560, acc1);
        path_wmma(P11, r2, x2v, n, kb, wz + 2816, acc2);
        path_wmma(P12, r2, x2v, n, kb, wz + 3072, acc0);
        path_wmma(P13, r2, x2v, n, kb, wz + 3328, acc1);
        path_wmma(P14, r2, x2v, n, kb, wz + 3584, acc2);
    }

    // Store: D VGPR r -> (wo = r + 8*(lane>=16), k = n); fold 1/sqrt(path count)
    float* oz = out + z * 144;
    const int wb = (lane >> 4) * 8;
    if (n == 0) {
#pragma unroll
        for (int r = 0; r < 8; ++r) oz[wb + r] = acc0[r] * 0.14433757f;            // 1/sqrt(48)
    }
    if (n < 3) {
#pragma unroll
        for (int r = 0; r < 8; ++r) oz[16 + (wb + r) * 3 + n] = acc1[r] * 0.10206207f; // 1/sqrt(96)
    }
    if (n < 5) {
#pragma unroll
        for (int r = 0; r < 8; ++r) oz[64 + (wb + r) * 5 + n] = acc2[r] * 0.10206207f; // 1/sqrt(96)
    }
}

extern "C" void kernel_launch(void* const* d_in, const int* in_sizes, int n_in,
                              void* d_out, int out_size, void* d_ws, size_t ws_size,
                              hipStream_t stream)
{
    const float* x1 = (const float*)d_in[0];  // [B,144]
    const float* x2 = (const float*)d_in[1];  // [B,9]
    const float* w  = (const float*)d_in[2];  // [B,3840]
    float* out = (float*)d_out;               // [B,144]
    const int batch  = in_sizes[0] / 144;
    const int blocks = (batch + 7) / 8;       // 8 elements (waves) per 256-thread block
    hipLaunchKernelGGL(wtp_kernel, dim3(blocks), dim3(256), 0, stream,
                       x1, x2, w, out, batch);
}